// STGNN_2869038154383
// MI455X (gfx1250) — compile-verified
//
#include <hip/hip_runtime.h>
#include <math.h>

// ---------------------------------------------------------------------------
// STGNN for MI455X (gfx1250, wave32). All heavy GEMMs use V_WMMA_F32_16X16X4_F32
// (fp32 matrix cores -> no precision loss). Shapes: B=32 T=12 N=207 E=1656
// HID=64 HEADS=4 F=16 TOUT=12. G=B*T=384, GN=G*N=79488, BN=B*N=6624.
// ---------------------------------------------------------------------------

#define Bdim 32
#define Tdim 12
#define Ndim 207
#define Edim 1656
#define HID 64
#define GNUM (Bdim * Tdim)          // 384
#define GN (GNUM * Ndim)            // 79488
#define BN (Bdim * Ndim)            // 6624
#define TOUT 12

typedef __attribute__((ext_vector_type(2))) float v2f;
typedef __attribute__((ext_vector_type(8))) float v8f;

__device__ __forceinline__ float sigf(float x) { return 1.0f / (1.0f + __expf(-x)); }
__device__ __forceinline__ float eluf(float x) { return x > 0.0f ? x : (__expf(x) - 1.0f); }
__device__ __forceinline__ float lreluf(float x) { return x > 0.0f ? x : 0.2f * x; }

#define WMMA_F32(A_, B_, C_) \
  __builtin_amdgcn_wmma_f32_16x16x4_f32(false, (A_), false, (B_), (short)0, (C_), false, false)

// ---------------------------------------------------------------------------
// fp32 WMMA GEMM specialized for Ncol = 64:
//   C[M,64] = rowmap(A)[M,64] @ B[64,64] (+bias)(+elu)
// One wave per 16-row stripe; 4 accumulators cover all 64 columns so each
// A fragment is reused 4x. K fully unrolled -> 64 chained WMMAs per wave.
// rowmap: 0 = identity; 2 = gather last timestep (src row = r*T + (T-1)).
// ---------------------------------------------------------------------------
__global__ __launch_bounds__(256) void gemm16x64_wmma(
    const float* __restrict__ A, const float* __restrict__ Bm,
    float* __restrict__ C, const float* __restrict__ bias,
    int M, int rowmap, int act)
{
  const int lane = threadIdx.x & 31;
  const int wave = blockIdx.x * (blockDim.x >> 5) + (threadIdx.x >> 5);
  if (wave >= (M >> 4)) return;            // wave-uniform
  const int mbase = wave << 4;
  const int mrow  = lane & 15;             // A-frag: lanes hold M=0..15 twice
  const int khalf = (lane >> 4) << 1;      // VGPR pair covers K, K+2 split
  const int ncol  = lane & 15;             // column within a 16-col tile

  int srow = mbase + mrow;
  if (rowmap == 2) srow = srow * Tdim + (Tdim - 1);
  const float* arow = A + (size_t)srow * HID + khalf;

  v8f acc0 = {0.f,0.f,0.f,0.f,0.f,0.f,0.f,0.f};
  v8f acc1 = acc0, acc2 = acc0, acc3 = acc0;

#pragma unroll
  for (int k0 = 0; k0 < HID; k0 += 4) {
    v2f a = *(const v2f*)(arow + k0);                 // global_load_b64
    const float* bp = Bm + (size_t)(k0 + khalf) * HID + ncol;
    v2f b0; b0.x = bp[0];  b0.y = bp[64];
    v2f b1; b1.x = bp[16]; b1.y = bp[80];
    v2f b2; b2.x = bp[32]; b2.y = bp[96];
    v2f b3; b3.x = bp[48]; b3.y = bp[112];
    acc0 = WMMA_F32(a, b0, acc0);
    acc1 = WMMA_F32(a, b1, acc1);
    acc2 = WMMA_F32(a, b2, acc2);
    acc3 = WMMA_F32(a, b3, acc3);
  }

#pragma unroll
  for (int tn = 0; tn < 4; ++tn) {
    v8f acc = tn == 0 ? acc0 : (tn == 1 ? acc1 : (tn == 2 ? acc2 : acc3));
    int col = tn * 16 + ncol;
    float bv = bias ? bias[col] : 0.0f;
#pragma unroll
    for (int j = 0; j < 8; ++j) {
      int row = mbase + j + ((lane >> 4) << 3);
      float v = acc[j] + bv;
      if (act == 1) v = eluf(v);
      C[(size_t)row * HID + col] = v;
    }
  }
}

// ---------------------------------------------------------------------------
// Projection: h0[gn, c] = x[gn] * proj_w[c] + proj_b[c]   (DIN == 1)
// ---------------------------------------------------------------------------
__global__ void proj_k(const float* __restrict__ x, const float* __restrict__ w,
                       const float* __restrict__ b, float* __restrict__ h0)
{
  int i = blockIdx.x * blockDim.x + threadIdx.x;
  if (i >= GN * HID) return;
  int gn = i >> 6, c = i & 63;
  h0[i] = x[gn] * w[c] + b[c];
}

// ---------------------------------------------------------------------------
// Deterministic dst-CSR (edge structure shared by both GAT blocks).
// Single block; stable single-thread scatter keeps FP sum order reproducible.
// ---------------------------------------------------------------------------
__global__ void build_csr(const int* __restrict__ ei, int* __restrict__ rowptr,
                          int* __restrict__ colidx)
{
  __shared__ int cnt[Ndim + 1];
  __shared__ int fill[Ndim + 1];
  const int* dst = ei + Edim;
  int tid = threadIdx.x;
  for (int i = tid; i <= Ndim; i += blockDim.x) cnt[i] = 0;
  __syncthreads();
  for (int e = tid; e < Edim; e += blockDim.x) atomicAdd(&cnt[dst[e] + 1], 1);
  __syncthreads();
  if (tid == 0) {
    for (int i = 1; i <= Ndim; ++i) cnt[i] += cnt[i - 1];
    for (int i = 0; i <= Ndim; ++i) { rowptr[i] = cnt[i]; fill[i] = cnt[i]; }
    for (int e = 0; e < Edim; ++e) { int d = dst[e]; colidx[fill[d]++] = e; }
  }
}

// ---------------------------------------------------------------------------
// Per (gn, head) attention pre-scores: s = <h2[gn, h*16:], a[h, :]>
// ---------------------------------------------------------------------------
__global__ void gat_scores(const float* __restrict__ h2,
                           const float* __restrict__ asrc,
                           const float* __restrict__ adst,
                           float* __restrict__ ssrc, float* __restrict__ sdst)
{
  int i = blockIdx.x * blockDim.x + threadIdx.x;
  if (i >= GN * 4) return;
  int gn = i >> 2, h = i & 3;
  const float* hp = h2 + (size_t)gn * HID + h * 16;
  float s1 = 0.f, s2 = 0.f;
#pragma unroll
  for (int f = 0; f < 16; ++f) {
    float v = hp[f];
    s1 += v * asrc[h * 16 + f];
    s2 += v * adst[h * 16 + f];
  }
  ssrc[i] = s1; sdst[i] = s2;
}

// ---------------------------------------------------------------------------
// Fused GAT: per-node edge softmax + weighted aggregate + bias + LN + ELU.
// One wave per (graph, node); 64 channels as 2 per lane; shuffle reductions.
// ---------------------------------------------------------------------------
__global__ __launch_bounds__(256) void gat_agg(
    const float* __restrict__ h2, const float* __restrict__ ssrc,
    const float* __restrict__ sdst, const int* __restrict__ rowptr,
    const int* __restrict__ colidx, const int* __restrict__ esrc,
    const float* __restrict__ ew, const float* __restrict__ bias,
    const float* __restrict__ lng, const float* __restrict__ lnb,
    float* __restrict__ outp)
{
  int wid = blockIdx.x * (blockDim.x >> 5) + (threadIdx.x >> 5);
  int lane = threadIdx.x & 31;
  if (wid >= GN) return;
  int g = wid / Ndim, n = wid % Ndim;
  int beg = rowptr[n], end = rowptr[n + 1];
  const float* ss = ssrc + (size_t)g * Ndim * 4;
  const float* sd = sdst + (size_t)g * Ndim * 4;
  float sd0 = sd[n * 4 + 0], sd1 = sd[n * 4 + 1];
  float sd2 = sd[n * 4 + 2], sd3 = sd[n * 4 + 3];

  float mx0 = -1e30f, mx1 = -1e30f, mx2 = -1e30f, mx3 = -1e30f;
  for (int p = beg; p < end; ++p) {
    int e = colidx[p]; int s = esrc[e];
    mx0 = fmaxf(mx0, lreluf(ss[s * 4 + 0] + sd0));
    mx1 = fmaxf(mx1, lreluf(ss[s * 4 + 1] + sd1));
    mx2 = fmaxf(mx2, lreluf(ss[s * 4 + 2] + sd2));
    mx3 = fmaxf(mx3, lreluf(ss[s * 4 + 3] + sd3));
  }
  float dn0 = 0.f, dn1 = 0.f, dn2 = 0.f, dn3 = 0.f;
  for (int p = beg; p < end; ++p) {
    int e = colidx[p]; int s = esrc[e];
    dn0 += __expf(lreluf(ss[s * 4 + 0] + sd0) - mx0);
    dn1 += __expf(lreluf(ss[s * 4 + 1] + sd1) - mx1);
    dn2 += __expf(lreluf(ss[s * 4 + 2] + sd2) - mx2);
    dn3 += __expf(lreluf(ss[s * 4 + 3] + sd3) - mx3);
  }
  // this lane's two channels: c0 = lane (heads 0/1), c1 = lane+32 (heads 2/3)
  int hA = lane >> 4;            // 0 or 1
  int hB = hA + 2;               // 2 or 3
  float mxA = hA ? mx1 : mx0, dnA = hA ? dn1 : dn0, sdA = hA ? sd1 : sd0;
  float mxB = hA ? mx3 : mx2, dnB = hA ? dn3 : dn2, sdB = hA ? sd3 : sd2;
  float invA = 1.0f / (dnA + 1e-16f);
  float invB = 1.0f / (dnB + 1e-16f);

  float a0 = 0.f, a1 = 0.f;
  for (int p = beg; p < end; ++p) {
    int e = colidx[p]; int s = esrc[e];
    float w = ew[e];
    const float* hrow = h2 + ((size_t)g * Ndim + s) * HID;
    float al0 = __expf(lreluf(ss[s * 4 + hA] + sdA) - mxA) * invA * w;
    float al1 = __expf(lreluf(ss[s * 4 + hB] + sdB) - mxB) * invB * w;
    a0 += al0 * hrow[lane];
    a1 += al1 * hrow[lane + 32];
  }
  a0 += bias[lane];
  a1 += bias[lane + 32];

  // LayerNorm over 64 channels (wave32 shuffle tree) + ELU
  float sum = a0 + a1;
#pragma unroll
  for (int m = 16; m >= 1; m >>= 1) sum += __shfl_xor(sum, m, 32);
  float mean = sum * (1.0f / 64.0f);
  float d0 = a0 - mean, d1 = a1 - mean;
  float vs = d0 * d0 + d1 * d1;
#pragma unroll
  for (int m = 16; m >= 1; m >>= 1) vs += __shfl_xor(vs, m, 32);
  float rstd = rsqrtf(vs * (1.0f / 64.0f) + 1e-5f);
  float y0 = eluf(d0 * rstd * lng[lane] + lnb[lane]);
  float y1 = eluf(d1 * rstd * lng[lane + 32] + lnb[lane + 32]);
  float* op = outp + ((size_t)g * Ndim + n) * HID;
  op[lane] = y0;
  op[lane + 32] = y1;
}

// ---------------------------------------------------------------------------
// Fused LSTM layer (both x-part and h-part GEMMs on WMMA).
// Block = 512 threads = 16 waves handles 16 sequences for all 12 steps.
// Wave w owns gate columns [16w, 16w+16). Step-invariant B fragments of
// wih^T / whh^T live in registers; h, c, x, gates live in LDS.
// remap=1: xin is GAT layout [B*T, N, 64]; remap=0: xin is [B*N, T, 64].
// ---------------------------------------------------------------------------
__global__ __launch_bounds__(512) void lstm_layer_k(
    const float* __restrict__ xin, const float* __restrict__ wih,
    const float* __restrict__ whh, const float* __restrict__ bih,
    const float* __restrict__ bhh, float* __restrict__ hout, int remap)
{
  __shared__ float hL[16][HID];
  __shared__ float cL[16][HID];
  __shared__ float xb[16][HID];
  __shared__ float gt[16][4 * HID];

  const int tid = threadIdx.x;
  const int lane = tid & 31;
  const int w = tid >> 5;                  // wave id -> column tile
  const int seqbase = blockIdx.x << 4;
  const int ncol = (w << 4) + (lane & 15); // gate column 0..255
  const int khalf = (lane >> 4) << 1;
  const int mrow = lane & 15;

  // Preload step-invariant B fragments: B[k][n] = W[ncol][k] (W is [256,64])
  v2f bh[16], bx[16];
#pragma unroll
  for (int kk = 0; kk < 16; ++kk) {
    int k = kk * 4 + khalf;
    bh[kk] = *(const v2f*)(whh + (size_t)ncol * HID + k);
    bx[kk] = *(const v2f*)(wih + (size_t)ncol * HID + k);
  }
  const float bias = bih[ncol] + bhh[ncol];

  for (int i = tid; i < 16 * HID; i += 512) {
    hL[i >> 6][i & 63] = 0.0f;
    cL[i >> 6][i & 63] = 0.0f;
  }
  __syncthreads();

  for (int t = 0; t < Tdim; ++t) {
    // stage x_t for the 16 sequences
    for (int i = tid; i < 16 * HID; i += 512) {
      int m = i >> 6, c = i & 63;
      int s = seqbase + m;
      size_t src;
      if (remap) {
        int b = s / Ndim, n = s % Ndim;
        src = ((size_t)(b * Tdim + t) * Ndim + n) * HID + c;
      } else {
        src = ((size_t)s * Tdim + t) * HID + c;
      }
      xb[m][c] = xin[src];
    }
    __syncthreads();

    v8f acc = {0.f,0.f,0.f,0.f,0.f,0.f,0.f,0.f};
#pragma unroll
    for (int kk = 0; kk < 16; ++kk) {
      int k = kk * 4 + khalf;
      v2f ah = *(const v2f*)(&hL[mrow][k]);   // ds_load_b64
      v2f ax = *(const v2f*)(&xb[mrow][k]);
      acc = WMMA_F32(ah, bh[kk], acc);
      acc = WMMA_F32(ax, bx[kk], acc);
    }
#pragma unroll
    for (int j = 0; j < 8; ++j) {
      int m = j + ((lane >> 4) << 3);
      gt[m][ncol] = acc[j] + bias;
    }
    __syncthreads();

    // gate nonlinearity + state update + stream h to global
    for (int i = tid; i < 16 * HID; i += 512) {
      int m = i >> 6, ch = i & 63;
      float ig = gt[m][ch];
      float fg = gt[m][HID + ch];
      float gg = gt[m][2 * HID + ch];
      float og = gt[m][3 * HID + ch];
      float c = sigf(fg) * cL[m][ch] + sigf(ig) * tanhf(gg);
      float h = sigf(og) * tanhf(c);
      cL[m][ch] = c;
      hL[m][ch] = h;
      hout[((size_t)(seqbase + m) * Tdim + t) * HID + ch] = h;
    }
    __syncthreads();
  }
}

// ---------------------------------------------------------------------------
// Attention core (only query row t = T-1 feeds the output slice z[:, -1]).
// One block (64 threads) per sequence: scores -> softmax -> ctx row 11.
// ---------------------------------------------------------------------------
__global__ __launch_bounds__(64) void attn_core(
    const float* __restrict__ q11, const float* __restrict__ k,
    const float* __restrict__ v, float* __restrict__ ctx11)
{
  __shared__ float sc[4][Tdim];
  const int s = blockIdx.x, tid = threadIdx.x;
  const float* qp = q11 + (size_t)s * HID;
  if (tid < 4 * Tdim) {
    int h = tid / Tdim, kt = tid % Tdim;
    const float* kp = k + ((size_t)s * Tdim + kt) * HID + h * 16;
    float d = 0.f;
#pragma unroll
    for (int f = 0; f < 16; ++f) d += qp[h * 16 + f] * kp[f];
    sc[h][kt] = d * 0.25f;                 // 1/sqrt(AF), AF=16
  }
  __syncthreads();
  if (tid < 4) {
    int h = tid;
    float m = sc[h][0];
    for (int i = 1; i < Tdim; ++i) m = fmaxf(m, sc[h][i]);
    float su = 0.f;
    for (int i = 0; i < Tdim; ++i) { float e = __expf(sc[h][i] - m); sc[h][i] = e; su += e; }
    float inv = 1.0f / su;
    for (int i = 0; i < Tdim; ++i) sc[h][i] *= inv;
  }
  __syncthreads();
  int h = tid >> 4;
  float acc = 0.f;
#pragma unroll
  for (int kt = 0; kt < Tdim; ++kt)
    acc += sc[h][kt] * v[((size_t)s * Tdim + kt) * HID + tid];
  ctx11[(size_t)s * HID + tid] = acc;
}

// ---------------------------------------------------------------------------
// Residual + LayerNorm3 on the t = T-1 slice. One wave per row.
// ---------------------------------------------------------------------------
__global__ __launch_bounds__(256) void ln_res(
    const float* __restrict__ ht2, const float* __restrict__ o11,
    const float* __restrict__ g, const float* __restrict__ b,
    float* __restrict__ z)
{
  int wid = blockIdx.x * (blockDim.x >> 5) + (threadIdx.x >> 5);
  int lane = threadIdx.x & 31;
  if (wid >= BN) return;
  const float* hp = ht2 + ((size_t)wid * Tdim + (Tdim - 1)) * HID;
  const float* op = o11 + (size_t)wid * HID;
  float x0 = hp[lane] + op[lane];
  float x1 = hp[lane + 32] + op[lane + 32];
  float sum = x0 + x1;
#pragma unroll
  for (int m = 16; m >= 1; m >>= 1) sum += __shfl_xor(sum, m, 32);
  float mean = sum * (1.0f / 64.0f);
  float d0 = x0 - mean, d1 = x1 - mean;
  float vs = d0 * d0 + d1 * d1;
#pragma unroll
  for (int m = 16; m >= 1; m >>= 1) vs += __shfl_xor(vs, m, 32);
  float rstd = rsqrtf(vs * (1.0f / 64.0f) + 1e-5f);
  float* zp = z + (size_t)wid * HID;
  zp[lane] = d0 * rstd * g[lane] + b[lane];
  zp[lane + 32] = d1 * rstd * g[lane + 32] + b[lane + 32];
}

// ---------------------------------------------------------------------------
// Final head: out[b, tout, n] = <a1[b*N+n, :], h2_w[:, tout]> + h2_b[tout]
// ---------------------------------------------------------------------------
__global__ void head2_k(const float* __restrict__ a1, const float* __restrict__ w,
                        const float* __restrict__ b, float* __restrict__ out)
{
  int i = blockIdx.x * blockDim.x + threadIdx.x;
  if (i >= BN * TOUT) return;
  int s = i / TOUT, to = i % TOUT;
  int bb = s / Ndim, n = s % Ndim;
  const float* ap = a1 + (size_t)s * HID;
  float acc = b[to];
#pragma unroll
  for (int c = 0; c < HID; ++c) acc += ap[c] * w[c * TOUT + to];
  out[((size_t)bb * TOUT + to) * Ndim + n] = acc;
}

// ---------------------------------------------------------------------------
// Host-side launcher
// ---------------------------------------------------------------------------
extern "C" void kernel_launch(void* const* d_in, const int* in_sizes, int n_in,
                              void* d_out, int out_size, void* d_ws, size_t ws_size,
                              hipStream_t stream)
{
  const float* x       = (const float*)d_in[0];
  const int*   ei      = (const int*)d_in[1];
  const float* ew      = (const float*)d_in[2];
  const float* proj_w  = (const float*)d_in[3];
  const float* proj_b  = (const float*)d_in[4];
  const float* g1_w    = (const float*)d_in[5];
  const float* g1_asrc = (const float*)d_in[6];
  const float* g1_adst = (const float*)d_in[7];
  const float* g1_b    = (const float*)d_in[8];
  const float* ln1_g   = (const float*)d_in[9];
  const float* ln1_b   = (const float*)d_in[10];
  const float* g2_w    = (const float*)d_in[11];
  const float* g2_asrc = (const float*)d_in[12];
  const float* g2_adst = (const float*)d_in[13];
  const float* g2_b    = (const float*)d_in[14];
  const float* ln2_g   = (const float*)d_in[15];
  const float* ln2_b   = (const float*)d_in[16];
  const float* l0_wih  = (const float*)d_in[17];
  const float* l0_whh  = (const float*)d_in[18];
  const float* l0_bih  = (const float*)d_in[19];
  const float* l0_bhh  = (const float*)d_in[20];
  const float* l1_wih  = (const float*)d_in[21];
  const float* l1_whh  = (const float*)d_in[22];
  const float* l1_bih  = (const float*)d_in[23];
  const float* l1_bhh  = (const float*)d_in[24];
  const float* wq      = (const float*)d_in[25];
  const float* wk      = (const float*)d_in[26];
  const float* wv      = (const float*)d_in[27];
  const float* wo      = (const float*)d_in[28];
  const float* ln3_g   = (const float*)d_in[29];
  const float* ln3_b   = (const float*)d_in[30];
  const float* h1_w    = (const float*)d_in[31];
  const float* h1_b    = (const float*)d_in[32];
  const float* h2_w    = (const float*)d_in[33];
  const float* h2_b    = (const float*)d_in[34];
  float* out = (float*)d_out;

  // Bump allocator over d_ws (256B aligned slices)
  char* base = (char*)d_ws;
  size_t off = 0;
  auto alloc = [&](size_t bytes) -> void* {
    void* p = base + off;
    off = (off + bytes + 255) & ~(size_t)255;
    return p;
  };
  const size_t BIGF = (size_t)GN * HID;          // 79488*64 floats
  float* bufA  = (float*)alloc(BIGF * 4);        // node features / ht2
  float* bufB  = (float*)alloc(BIGF * 4);        // gemm scratch / k
  float* bufC  = (float*)alloc(BIGF * 4);        // gat out / lstm0 out / v
  float* ssrc  = (float*)alloc((size_t)GN * 4 * 4);
  float* sdst  = (float*)alloc((size_t)GN * 4 * 4);
  float* q11   = (float*)alloc((size_t)BN * HID * 4);
  float* ctx11 = (float*)alloc((size_t)BN * HID * 4);
  float* o11   = (float*)alloc((size_t)BN * HID * 4);
  float* zb    = (float*)alloc((size_t)BN * HID * 4);
  float* a1    = (float*)alloc((size_t)BN * HID * 4);
  int* rowptr  = (int*)alloc((Ndim + 1) * 4);
  int* colidx  = (int*)alloc(Edim * 4);

  const int* esrc = ei;  // ei[0,:] = src, ei[1,:] = dst

  auto gemm = [&](const float* A, const float* Bm, float* C, const float* bias,
                  int M, int rowmap, int act) {
    int waves = M >> 4;                    // one wave per 16-row stripe
    int blocks = (waves + 7) / 8;
    gemm16x64_wmma<<<blocks, 256, 0, stream>>>(A, Bm, C, bias, M, rowmap, act);
  };

  // 0) edge CSR (deterministic) + projection
  build_csr<<<1, 256, 0, stream>>>(ei, rowptr, colidx);
  proj_k<<<(GN * HID + 255) / 256, 256, 0, stream>>>(x, proj_w, proj_b, bufA);

  // 1) GAT block 1: A -> C
  gemm(bufA, g1_w, bufB, nullptr, GN, 0, 0);
  gat_scores<<<(GN * 4 + 255) / 256, 256, 0, stream>>>(bufB, g1_asrc, g1_adst, ssrc, sdst);
  gat_agg<<<(GN + 7) / 8, 256, 0, stream>>>(bufB, ssrc, sdst, rowptr, colidx,
                                            esrc, ew, g1_b, ln1_g, ln1_b, bufC);

  // 2) GAT block 2: C -> A
  gemm(bufC, g2_w, bufB, nullptr, GN, 0, 0);
  gat_scores<<<(GN * 4 + 255) / 256, 256, 0, stream>>>(bufB, g2_asrc, g2_adst, ssrc, sdst);
  gat_agg<<<(GN + 7) / 8, 256, 0, stream>>>(bufB, ssrc, sdst, rowptr, colidx,
                                            esrc, ew, g2_b, ln2_g, ln2_b, bufA);

  // 3) LSTM layers (fused WMMA recurrence): A(remap) -> C -> A
  lstm_layer_k<<<BN / 16, 512, 0, stream>>>(bufA, l0_wih, l0_whh, l0_bih, l0_bhh, bufC, 1);
  lstm_layer_k<<<BN / 16, 512, 0, stream>>>(bufC, l1_wih, l1_whh, l1_bih, l1_bhh, bufA, 0);

  // 4) Temporal attention (only q at t=T-1 contributes to the output)
  gemm(bufA, wk, bufB, nullptr, GN, 0, 0);     // k, all t
  gemm(bufA, wv, bufC, nullptr, GN, 0, 0);     // v, all t
  gemm(bufA, wq, q11, nullptr, BN, 2, 0);      // q, t=T-1 slice
  attn_core<<<BN, 64, 0, stream>>>(q11, bufB, bufC, ctx11);
  gemm(ctx11, wo, o11, nullptr, BN, 0, 0);     // ctx @ wo

  // 5) residual + LN3 + head MLP
  ln_res<<<BN / 8, 256, 0, stream>>>(bufA, o11, ln3_g, ln3_b, zb);
  gemm(zb, h1_w, a1, h1_b, BN, 0, 1);          // elu(z@h1 + b1)
  head2_k<<<(BN * TOUT + 255) / 256, 256, 0, stream>>>(a1, h2_w, h2_b, out);
}